// GlobalAttention_48670569398523
// MI455X (gfx1250) — compile-verified
//
#include <hip/hip_runtime.h>
#include <math.h>

// Problem dims (compile-time, match reference)
#define DDIM 1024
#define BDIM 32
#define SDIM 4096
#define NCHUNKS 32
#define SCHUNK (SDIM / NCHUNKS)   // 128 rows per (batch,chunk) block
#define ROWS 8                    // rows processed per inner iteration
#define TPB 256                   // threads per block in the fused pass (8 waves)

typedef __attribute__((ext_vector_type(2))) float v2f;
typedef __attribute__((ext_vector_type(8))) float v8f;

// ---------------------------------------------------------------------------
// K1: dec_att[m,n] = sum_k dec_output[m,k] * W_in[n,k]   (x @ W.T)
// fp32 WMMA 16x16x4. One wave per block; grid = (N/16, M/16).
// A 16x4 layout: lanes 0-15 hold row m=lane, VGPR0/1 = K0,K1; lanes 16-31 = K2,K3.
// B 4x16 layout mirrored (lane = column n, VGPRs = K within lane half).
// ---------------------------------------------------------------------------
__global__ void __launch_bounds__(32) linear_in_wmma(const float* __restrict__ X,
                                                     const float* __restrict__ W,
                                                     float* __restrict__ Y) {
  const int lane = threadIdx.x;
  const int n0 = blockIdx.x * 16;
  const int m0 = blockIdx.y * 16;
  const int half = lane >> 4;     // 0: K0,K1   1: K2,K3
  const int lid  = lane & 15;
  const int koff = half * 2;

  v8f c = {};
  const float* xrow = X + (m0 + lid) * DDIM + koff;   // A row (dec_output)
  const float* wrow = W + (n0 + lid) * DDIM + koff;   // B col n -> W_in row n
  for (int k = 0; k < DDIM; k += 4) {
    v2f a = *(const v2f*)(xrow + k);
    v2f b = *(const v2f*)(wrow + k);
    c = __builtin_amdgcn_wmma_f32_16x16x4_f32(false, a, false, b, (short)0, c,
                                              false, false);
  }
  // C/D layout: lanes 0-15 -> M = vgpr index, N = lane; lanes 16-31 -> M = 8+vgpr.
  const int n  = n0 + lid;
  const int mb = m0 + half * 8;
#pragma unroll
  for (int r = 0; r < 8; ++r) Y[(mb + r) * DDIM + n] = c[r];
}

// ---------------------------------------------------------------------------
// K2: fused single-pass over enc_outputs (online softmax, flash style).
// Grid: (NCHUNKS, B). Each block handles SCHUNK rows of one batch.
// Each of 256 threads owns 4 contiguous dims (float4). Each enc element is
// loaded from HBM exactly once and used for both the score dot product and
// the weighted accumulation.
// Outputs per (b,chunk): raw scores, partial m, partial l, partial acc[D].
// ---------------------------------------------------------------------------
__global__ void __launch_bounds__(TPB) attn_fused(
    const float* __restrict__ enc,      // [B,S,D]
    const float* __restrict__ dec_att,  // [B,D]
    float* __restrict__ scores,         // [B,S]  raw scores
    float* __restrict__ part_m,         // [B,NCHUNKS]
    float* __restrict__ part_l,         // [B,NCHUNKS]
    float* __restrict__ part_acc)       // [B,NCHUNKS,D]
{
  const int b     = blockIdx.y;
  const int chunk = blockIdx.x;
  const int t     = threadIdx.x;
  const int lane  = t & 31;
  const int wave  = t >> 5;           // 0..7
  const int d0    = t * 4;

  __shared__ float red[8][ROWS];
  __shared__ float sc[ROWS];

  const float4 da = *(const float4*)(dec_att + b * DDIM + d0);

  float  m = -INFINITY;
  float  l = 0.0f;
  float4 acc; acc.x = acc.y = acc.z = acc.w = 0.0f;

  const int    srow0 = chunk * SCHUNK;
  const float* encb  = enc + (size_t)b * SDIM * DDIM;

  for (int it = 0; it < SCHUNK; it += ROWS) {
    float4 e[ROWS];
    float  p[ROWS];
#pragma unroll
    for (int r = 0; r < ROWS; ++r) {
      const float* rowp = encb + (size_t)(srow0 + it + r) * DDIM + d0;
      e[r] = *(const float4*)rowp;
      if (it + ROWS < SCHUNK)                      // prefetch next row group
        __builtin_prefetch(rowp + (size_t)ROWS * DDIM, 0, 0);
      p[r] = e[r].x * da.x + e[r].y * da.y + e[r].z * da.z + e[r].w * da.w;
    }
    // wave-level reduction of each row's partial dot
#pragma unroll
    for (int r = 0; r < ROWS; ++r) {
      float v = p[r];
      for (int off = 16; off > 0; off >>= 1) v += __shfl_down(v, off, 32);
      if (lane == 0) red[wave][r] = v;
    }
    __syncthreads();
    if (t < ROWS) {
      float v = 0.0f;
      for (int w = 0; w < 8; ++w) v += red[w][t];
      sc[t] = v;
      scores[b * SDIM + srow0 + it + t] = v;       // raw score for output pass
    }
    __syncthreads();
    // uniform online-softmax update; each thread updates its 4 dims
#pragma unroll
    for (int r = 0; r < ROWS; ++r) {
      const float s_r   = sc[r];
      const float m_new = fmaxf(m, s_r);
      const float corr  = __expf(m - m_new);
      const float w     = __expf(s_r - m_new);
      l = l * corr + w;
      acc.x = acc.x * corr + w * e[r].x;
      acc.y = acc.y * corr + w * e[r].y;
      acc.z = acc.z * corr + w * e[r].z;
      acc.w = acc.w * corr + w * e[r].w;
      m = m_new;
    }
    __syncthreads();   // protect red/sc before next iteration's writes
  }

  *(float4*)(part_acc + ((size_t)b * NCHUNKS + chunk) * DDIM + d0) = acc;
  if (t == 0) {
    part_m[b * NCHUNKS + chunk] = m;
    part_l[b * NCHUNKS + chunk] = l;
  }
}

// ---------------------------------------------------------------------------
// K3: combine chunk partials -> global M,L and weighted[b,:] (normalized).
// Grid: B blocks of 256 threads (thread owns 4 dims).
// ---------------------------------------------------------------------------
__global__ void __launch_bounds__(TPB) reduce_partials(
    const float* __restrict__ part_m, const float* __restrict__ part_l,
    const float* __restrict__ part_acc,
    float* __restrict__ weighted,   // [B,D]
    float* __restrict__ Mb, float* __restrict__ Lb)
{
  const int b  = blockIdx.x;
  const int t  = threadIdx.x;
  const int d0 = t * 4;

  float M = -INFINITY;
  for (int c = 0; c < NCHUNKS; ++c) M = fmaxf(M, part_m[b * NCHUNKS + c]);
  float L = 0.0f;
  for (int c = 0; c < NCHUNKS; ++c)
    L += part_l[b * NCHUNKS + c] * __expf(part_m[b * NCHUNKS + c] - M);

  float4 acc; acc.x = acc.y = acc.z = acc.w = 0.0f;
  for (int c = 0; c < NCHUNKS; ++c) {
    const float s = __expf(part_m[b * NCHUNKS + c] - M);
    const float4 pa = *(const float4*)(part_acc + ((size_t)b * NCHUNKS + c) * DDIM + d0);
    acc.x += s * pa.x; acc.y += s * pa.y; acc.z += s * pa.z; acc.w += s * pa.w;
  }
  const float inv = 1.0f / L;
  float4 o; o.x = acc.x * inv; o.y = acc.y * inv; o.z = acc.z * inv; o.w = acc.w * inv;
  *(float4*)(weighted + b * DDIM + d0) = o;
  if (t == 0) { Mb[b] = M; Lb[b] = L; }
}

// ---------------------------------------------------------------------------
// K4: weights output = exp(score - M_b) / L_b,  over [B,S]
// ---------------------------------------------------------------------------
__global__ void __launch_bounds__(TPB) normalize_weights(
    const float* __restrict__ scores, const float* __restrict__ Mb,
    const float* __restrict__ Lb, float* __restrict__ wout)
{
  const int i = blockIdx.x * blockDim.x + threadIdx.x;   // 0 .. B*S-1
  const int b = i / SDIM;
  wout[i] = __expf(scores[i] - Mb[b]) / Lb[b];
}

// ---------------------------------------------------------------------------
// K5: output[m,n] = tanh( sum_{k<2D} combined[m,k] * W_out[n,k] )
// combined = [weighted | dec_output]. fp32 WMMA, same tiling as K1.
// ---------------------------------------------------------------------------
__global__ void __launch_bounds__(32) linear_out_wmma(
    const float* __restrict__ weighted, const float* __restrict__ dec,
    const float* __restrict__ Wout,   // [D, 2D] row-major
    float* __restrict__ out)          // [B, D]
{
  const int lane = threadIdx.x;
  const int n0 = blockIdx.x * 16;
  const int m0 = blockIdx.y * 16;
  const int half = lane >> 4;
  const int lid  = lane & 15;
  const int koff = half * 2;

  v8f c = {};
  const float* w1 = Wout + (size_t)(n0 + lid) * (2 * DDIM) + koff;   // k < D half
  const float* a1 = weighted + (m0 + lid) * DDIM + koff;
  for (int k = 0; k < DDIM; k += 4) {
    v2f a = *(const v2f*)(a1 + k);
    v2f b = *(const v2f*)(w1 + k);
    c = __builtin_amdgcn_wmma_f32_16x16x4_f32(false, a, false, b, (short)0, c,
                                              false, false);
  }
  const float* a2 = dec + (m0 + lid) * DDIM + koff;                   // k >= D half
  const float* w2 = w1 + DDIM;
  for (int k = 0; k < DDIM; k += 4) {
    v2f a = *(const v2f*)(a2 + k);
    v2f b = *(const v2f*)(w2 + k);
    c = __builtin_amdgcn_wmma_f32_16x16x4_f32(false, a, false, b, (short)0, c,
                                              false, false);
  }
  const int n  = n0 + lid;
  const int mb = m0 + half * 8;
#pragma unroll
  for (int r = 0; r < 8; ++r) out[(mb + r) * DDIM + n] = tanhf(c[r]);
}

// ---------------------------------------------------------------------------
extern "C" void kernel_launch(void* const* d_in, const int* in_sizes, int n_in,
                              void* d_out, int out_size, void* d_ws, size_t ws_size,
                              hipStream_t stream) {
  const float* dec_output  = (const float*)d_in[0];  // [B, D]
  const float* enc_outputs = (const float*)d_in[1];  // [B, S, D]
  const float* W_in        = (const float*)d_in[2];  // [D, D]
  const float* W_out       = (const float*)d_in[3];  // [D, 2D]

  float* out_output  = (float*)d_out;                 // [B, D]
  float* out_weights = (float*)d_out + BDIM * DDIM;   // [B, S]

  // Workspace carve-up (floats); total ~1.25M floats (~4.8 MB)
  float* ws       = (float*)d_ws;
  float* dec_att  = ws;                                   // B*D      = 32768
  float* scores   = dec_att + BDIM * DDIM;                // B*S      = 131072
  float* part_m   = scores + BDIM * SDIM;                 // B*NC     = 1024
  float* part_l   = part_m + BDIM * NCHUNKS;              // B*NC     = 1024
  float* Mb       = part_l + BDIM * NCHUNKS;              // B        = 32
  float* Lb       = Mb + BDIM;                            // B        = 32
  float* weighted = Lb + BDIM;                            // B*D      = 32768
  float* part_acc = weighted + BDIM * DDIM;               // B*NC*D   = 1048576

  // K1: dec_att = dec_output @ W_in.T   (fp32 WMMA)
  linear_in_wmma<<<dim3(DDIM / 16, BDIM / 16), dim3(32), 0, stream>>>(
      dec_output, W_in, dec_att);

  // K2: fused online-softmax pass over enc_outputs (single HBM sweep)
  attn_fused<<<dim3(NCHUNKS, BDIM), dim3(TPB), 0, stream>>>(
      enc_outputs, dec_att, scores, part_m, part_l, part_acc);

  // K3: combine chunk partials -> M, L, weighted
  reduce_partials<<<dim3(BDIM), dim3(TPB), 0, stream>>>(
      part_m, part_l, part_acc, weighted, Mb, Lb);

  // K4: weights output
  normalize_weights<<<dim3((BDIM * SDIM) / TPB), dim3(TPB), 0, stream>>>(
      scores, Mb, Lb, out_weights);

  // K5: output = tanh([weighted | dec_output] @ W_out.T)   (fp32 WMMA)
  linear_out_wmma<<<dim3(DDIM / 16, BDIM / 16), dim3(32), 0, stream>>>(
      weighted, dec_output, W_out, out_output);
}